// MetricSelfAttention_36593121362164
// MI455X (gfx1250) — compile-verified
//
#include <hip/hip_runtime.h>
#include <hip/hip_bf16.h>

// MetricSelfAttention for MI455X (gfx1250), fp32 WMMA path.
//
// Algebraic reduction: nudged = (p1 M p2^T) p1 = p1 (M (p2^T p1)).
// S[b,n] = p2^T p1 (64x64), G = M S, O = p1 G. This removes the
// [B,N,2048,2048] intermediate entirely (268MB + 17 GFLOP saved).
// Problem is HBM-bound (~60MB @ 23.3TB/s), so precision stays fp32 and the
// matrix ops use V_WMMA_F32_16X16X4_F32. Global->LDS staging uses the CDNA5
// async-DMA path (GLOBAL_LOAD_ASYNC_TO_LDS_*, ASYNCcnt) with double
// buffering so tile DMA overlaps the WMMA k-loop.

typedef float v2f __attribute__((ext_vector_type(2)));
typedef float v8f __attribute__((ext_vector_type(8)));

#define B_DIM   2
#define W_DIM   2048
#define C_DIM   512
#define N_HEADS 8
#define K_DIM   64
#define HALF_N  2016   // 64*63/2

// ---- CDNA5 async global->LDS helpers (ASYNCcnt-tracked DMA) -------------
__device__ __forceinline__ unsigned lds_off(const void* p) {
  // LDS aperture: generic addr[31:0] is the LDS byte offset.
  return (unsigned)(unsigned long long)p;
}
__device__ __forceinline__ void async_ld_b128(unsigned lds, const float* g) {
  asm volatile("global_load_async_to_lds_b128 %0, %1, off" :: "v"(lds), "v"(g) : "memory");
}
__device__ __forceinline__ void async_ld_b32(unsigned lds, const float* g) {
  asm volatile("global_load_async_to_lds_b32 %0, %1, off" :: "v"(lds), "v"(g) : "memory");
}
template <int N>
__device__ __forceinline__ void wait_async() {
  asm volatile("s_wait_asynccnt %0" :: "n"(N) : "memory");
}

// ---------------------------------------------------------------- LayerNorm
__global__ __launch_bounds__(256) void ln_kernel(
    const float* __restrict__ x1, const float* __restrict__ x2,
    const float* __restrict__ gamma, const float* __restrict__ beta,
    float* __restrict__ y1, float* __restrict__ y2)
{
  const int row = blockIdx.x;
  const int t   = threadIdx.x;
  const float* src = blockIdx.y ? x2 : x1;
  float*       dst = blockIdx.y ? y2 : y1;
  const float v0 = src[(size_t)row * C_DIM + t];
  const float v1 = src[(size_t)row * C_DIM + t + 256];
  __shared__ float rs[256], rq[256];
  rs[t] = v0 + v1;
  rq[t] = v0 * v0 + v1 * v1;
  __syncthreads();
  for (int off = 128; off > 0; off >>= 1) {
    if (t < off) { rs[t] += rs[t + off]; rq[t] += rq[t + off]; }
    __syncthreads();
  }
  const float mean = rs[0] * (1.0f / C_DIM);
  const float var  = rq[0] * (1.0f / C_DIM) - mean * mean;
  const float rstd = rsqrtf(var + 1e-5f);
  dst[(size_t)row * C_DIM + t]       = (v0 - mean) * rstd * gamma[t]       + beta[t];
  dst[(size_t)row * C_DIM + t + 256] = (v1 - mean) * rstd * gamma[t + 256] + beta[t + 256];
}

// ---------------------------------------------------------------- WMMA GEMM
// Block tile 128x64, BK=32, 8 waves; each wave computes a 32x32 tile (2x2
// WMMA 16x16 accumulators). Double-buffered LDS tiles fed by async DMA.
// Row strides: As 36 floats (16B-aligned rows for b128 async; banks 36l%64
// all distinct and even, hi-halfgroup +2 disjoint), Bs 72 floats (halfgroup
// rows k/k+2 land 16 banks apart).
#define BM 128
#define BN 64
#define BK 32
#define LDA_S 36
#define LDB_S 72

// MODE: 0 = B row-major (ldb), 1 = proj_nck [N,C,K], 2 = W_mixer transposed
template <int MODE>
__global__ __launch_bounds__(256) void gemm_wmma(
    const float* __restrict__ A, int lda, long AzH, long AzL,
    const float* __restrict__ B, int ldb, long BzH, long BzL,
    float* __restrict__ Cout, int ldc, long CzH, long CzL,
    int Kc, const float* __restrict__ bias)
{
  __shared__ float As[2][BM * LDA_S];
  __shared__ float Bs[2][BK * LDB_S];
  const int zz = blockIdx.z, zh = zz >> 3, zl = zz & 7;
  A    += (size_t)zh * AzH + (size_t)zl * AzL;
  B    += (size_t)zh * BzH + (size_t)zl * BzL;
  Cout += (size_t)zh * CzH + (size_t)zl * CzL;
  const int bm = blockIdx.x * BM, bn = blockIdx.y * BN;
  const int t = threadIdx.x, lane = t & 31, wv = t >> 5;
  const int wmo = (wv & 3) * 32, wno = (wv >> 2) * 32;
  const int lm = lane & 15, lhi = lane >> 4;

  // 6 async-DMA instructions per thread per chunk (4x A.b128 + 2x B).
  auto issue_tile = [&](int buf, int kc0) {
    #pragma unroll
    for (int p = 0; p < 4; ++p) {
      const int row = (t >> 3) + p * 32;
      const int c4  = (t & 7) * 4;
      async_ld_b128(lds_off(&As[buf][row * LDA_S + c4]),
                    A + (size_t)(bm + row) * lda + kc0 + c4);
    }
    if (MODE == 0) {
      #pragma unroll
      for (int p = 0; p < 2; ++p) {
        const int e = t + p * 256, k = e >> 4, n4 = (e & 15) * 4;
        async_ld_b128(lds_off(&Bs[buf][k * LDB_S + n4]),
                      B + (size_t)(kc0 + k) * ldb + bn + n4);
      }
    } else if (MODE == 1) {  // proj_nck: elem(k,n) = proj[(n>>6)*C*K + k*K + (n&63)]
      #pragma unroll
      for (int p = 0; p < 2; ++p) {
        const int e = t + p * 256, k = e >> 4, n4 = (e & 15) * 4;
        const int ng = bn + n4;
        async_ld_b128(lds_off(&Bs[buf][k * LDB_S + n4]),
                      B + (size_t)(ng >> 6) * (C_DIM * K_DIM)
                        + (size_t)(kc0 + k) * K_DIM + (ng & 63));
      }
    } else {  // W_mixer^T: elem(k,n) = Wm[n*C + k]  (coalesce along k)
      #pragma unroll
      for (int p = 0; p < 2; ++p) {
        const int e = t + p * 256, n = e >> 5, k = e & 31;
        async_ld_b32(lds_off(&Bs[buf][k * LDB_S + n]),
                     B + (size_t)(bn + n) * C_DIM + kc0 + k);
      }
    }
  };

  const v8f vzero = {0, 0, 0, 0, 0, 0, 0, 0};
  v8f acc[2][2];
  #pragma unroll
  for (int i = 0; i < 2; ++i)
    #pragma unroll
    for (int j = 0; j < 2; ++j) acc[i][j] = vzero;

  const int nch = Kc / BK;
  issue_tile(0, 0);
  for (int c = 0; c < nch; ++c) {
    const int cur = c & 1;
    if (c + 1 < nch) {
      issue_tile(cur ^ 1, (c + 1) * BK);  // prefetch next chunk (other buffer)
      wait_async<6>();                    // in-order: my chunk-c DMAs landed
    } else {
      wait_async<0>();
    }
    __syncthreads();                      // all waves' chunk-c DMAs visible

    #pragma unroll
    for (int ks = 0; ks < 8; ++ks) {
      const int kb = ks * 4 + lhi * 2;    // per-lane K base (ISA A/B striping)
      v2f af[2], bf[2];
      #pragma unroll
      for (int mt = 0; mt < 2; ++mt) {
        const int mrow = wmo + mt * 16 + lm;
        af[mt] = *(const v2f*)&As[cur][mrow * LDA_S + kb];   // ds_load_b64
      }
      #pragma unroll
      for (int nt = 0; nt < 2; ++nt) {
        const int ncol = wno + nt * 16 + lm;
        bf[nt].x = Bs[cur][kb * LDB_S + ncol];
        bf[nt].y = Bs[cur][(kb + 1) * LDB_S + ncol];
      }
      #pragma unroll
      for (int mt = 0; mt < 2; ++mt)
        #pragma unroll
        for (int nt = 0; nt < 2; ++nt)
          acc[mt][nt] = __builtin_amdgcn_wmma_f32_16x16x4_f32(
              false, af[mt], false, bf[nt], (short)0, acc[mt][nt], false, false);
    }
    __syncthreads();                      // done reading buf `cur`; reusable
  }

  // --- store (C/D layout: VGPR r -> rows r / r+8, cols striped on lanes) ---
  #pragma unroll
  for (int mt = 0; mt < 2; ++mt) {
    #pragma unroll
    for (int nt = 0; nt < 2; ++nt) {
      const int col = bn + wno + nt * 16 + lm;
      const float bv = bias ? bias[col] : 0.0f;
      const int rowb = bm + wmo + mt * 16 + lhi * 8;
      #pragma unroll
      for (int r = 0; r < 8; ++r)
        Cout[(size_t)(rowb + r) * ldc + col] = acc[mt][nt][r] + bv;
    }
  }
}

// ------------------------------------------------- S = p2^T p1 (per b,n)
// grid (8 W-chunks, 16 bn). Async-DMA 64x64 p-tiles into LDS; each wave
// computes a 16x64 strip of S over half the chunk; partials merged with
// global f32 atomics into memset-zeroed S.
#define PS 68
__global__ __launch_bounds__(256) void s_partial(
    const float* __restrict__ p1, const float* __restrict__ p2, float* __restrict__ S)
{
  __shared__ float p1s[64 * PS];
  __shared__ float p2s[64 * PS];
  const int bn = blockIdx.y;               // b*8+n
  const int b = bn >> 3, n = bn & 7;
  const size_t base = (size_t)b * W_DIM * C_DIM + (size_t)n * K_DIM;
  const int wbase = blockIdx.x * 256;
  const int t = threadIdx.x, lane = t & 31, wv = t >> 5;
  const int strip = wv & 3, half = wv >> 2;
  const int lm = lane & 15, lhi = lane >> 4;

  const v8f vzero = {0, 0, 0, 0, 0, 0, 0, 0};
  v8f acc[4];
  #pragma unroll
  for (int i = 0; i < 4; ++i) acc[i] = vzero;

  for (int st = 0; st < 4; ++st) {
    const int w0 = wbase + st * 64;
    #pragma unroll
    for (int p = 0; p < 4; ++p) {
      const int e = t + p * 256, row = e >> 4, c4 = (e & 15) * 4;
      const size_t g = base + (size_t)(w0 + row) * C_DIM + c4;
      async_ld_b128(lds_off(&p1s[row * PS + c4]), p1 + g);
      async_ld_b128(lds_off(&p2s[row * PS + c4]), p2 + g);
    }
    wait_async<0>();
    __syncthreads();
    #pragma unroll
    for (int ks = 0; ks < 8; ++ks) {
      const int kb = half * 32 + ks * 4 + lhi * 2;
      v2f af;  // A[m][w] = p2s[w][m]  (transposed read from LDS)
      af.x = p2s[kb * PS + strip * 16 + lm];
      af.y = p2s[(kb + 1) * PS + strip * 16 + lm];
      #pragma unroll
      for (int nt = 0; nt < 4; ++nt) {
        v2f bf;
        bf.x = p1s[kb * PS + nt * 16 + lm];
        bf.y = p1s[(kb + 1) * PS + nt * 16 + lm];
        acc[nt] = __builtin_amdgcn_wmma_f32_16x16x4_f32(
            false, af, false, bf, (short)0, acc[nt], false, false);
      }
    }
    __syncthreads();
  }
  float* Sd = S + (size_t)bn * K_DIM * K_DIM;
  #pragma unroll
  for (int nt = 0; nt < 4; ++nt) {
    const int col  = nt * 16 + lm;
    const int rowb = strip * 16 + lhi * 8;
    #pragma unroll
    for (int r = 0; r < 8; ++r)
      atomicAdd(&Sd[(rowb + r) * K_DIM + col], acc[nt][r]);
  }
}

// ------------------------------------------------- G = M(n) * S(b,n)
__global__ __launch_bounds__(256) void metric_apply(
    const float* __restrict__ halves, const float* __restrict__ diagonals,
    const float* __restrict__ S, float* __restrict__ G)
{
  __shared__ float Ms[64 * 64];
  __shared__ float Ss[64 * PS];
  const int z = blockIdx.x;  // b*8+n
  const int n = z & 7;
  const int t = threadIdx.x;
  #pragma unroll
  for (int q = 0; q < 16; ++q) {
    const int e = t * 16 + q;
    const int i = e >> 6, j = e & 63;
    float v;
    if (i == j) v = diagonals[n * K_DIM + i];
    else {
      const int a = i < j ? i : j, c = i < j ? j : i;
      v = halves[n * HALF_N + a * (2 * K_DIM - a - 1) / 2 + (c - a - 1)];
    }
    Ms[e] = v;
  }
  #pragma unroll
  for (int p = 0; p < 16; ++p) {
    const int e = t + p * 256;
    Ss[(e >> 6) * PS + (e & 63)] = S[(size_t)z * 4096 + e];
  }
  __syncthreads();
  const int i = t >> 2, nb = (t & 3) * 16;
  float acc[16];
  #pragma unroll
  for (int q = 0; q < 16; ++q) acc[q] = 0.f;
  for (int j = 0; j < 64; ++j) {
    const float m = Ms[i * 64 + j];
    #pragma unroll
    for (int q = 0; q < 16; ++q) acc[q] += m * Ss[j * PS + nb + q];
  }
  #pragma unroll
  for (int q = 0; q < 16; ++q) G[(size_t)z * 4096 + i * 64 + nb + q] = acc[q];
}

// ---------------------------------------------------------------- launch
extern "C" void kernel_launch(void* const* d_in, const int* in_sizes, int n_in,
                              void* d_out, int out_size, void* d_ws, size_t ws_size,
                              hipStream_t stream)
{
  (void)in_sizes; (void)n_in; (void)out_size; (void)ws_size;
  const float* x1        = (const float*)d_in[0];
  const float* x2        = (const float*)d_in[1];
  const float* gamma     = (const float*)d_in[2];
  const float* beta      = (const float*)d_in[3];
  const float* proj      = (const float*)d_in[4];
  const float* halves    = (const float*)d_in[5];
  const float* diagonals = (const float*)d_in[6];
  const float* Wm        = (const float*)d_in[7];
  const float* bmix      = (const float*)d_in[8];
  float* out = (float*)d_out;

  const size_t TOK = (size_t)B_DIM * W_DIM * C_DIM;  // 2,097,152 floats
  float* ws    = (float*)d_ws;
  float* x1n   = ws;
  float* x2n   = x1n + TOK;
  float* p1    = x2n + TOK;
  float* p2    = p1 + TOK;
  float* Sbuf  = p2 + TOK;             // 16 * 64 * 64
  float* Gbuf  = Sbuf + 16 * 4096;
  float* nudged = Gbuf + 16 * 4096;    // total ~42.5 MB of d_ws

  hipMemsetAsync(Sbuf, 0, 16 * 4096 * sizeof(float), stream);

  // LayerNorm both sequences
  ln_kernel<<<dim3(B_DIM * W_DIM, 2), 256, 0, stream>>>(x1, x2, gamma, beta, x1n, x2n);

  // p = xn @ proj  ([4096,512] x [512,512], per-head columns)
  gemm_wmma<1><<<dim3(32, 8, 1), 256, 0, stream>>>(
      x1n, C_DIM, 0, 0, proj, 0, 0, 0, p1, C_DIM, 0, 0, C_DIM, nullptr);
  gemm_wmma<1><<<dim3(32, 8, 1), 256, 0, stream>>>(
      x2n, C_DIM, 0, 0, proj, 0, 0, 0, p2, C_DIM, 0, 0, C_DIM, nullptr);

  // S = p2^T p1 per (b,n)
  s_partial<<<dim3(8, 16), 256, 0, stream>>>(p1, p2, Sbuf);

  // G = M S (builds symmetric metric on the fly)
  metric_apply<<<dim3(16), 256, 0, stream>>>(halves, diagonals, Sbuf, Gbuf);

  // O = p1 @ G per (b,n), written directly into [B,W,C] layout
  gemm_wmma<0><<<dim3(16, 1, 16), 256, 0, stream>>>(
      p1, C_DIM, (long)W_DIM * C_DIM, K_DIM,
      Gbuf, K_DIM, (long)8 * 4096, 4096,
      nudged, C_DIM, (long)W_DIM * C_DIM, K_DIM,
      K_DIM, nullptr);

  // out = nudged @ W_mixer^T + b_mixer
  gemm_wmma<2><<<dim3(32, 8, 1), 256, 0, stream>>>(
      nudged, C_DIM, 0, 0, Wm, 0, 0, 0, out, C_DIM, 0, 0, C_DIM, bmix);
}